// GraphCNN_64226940944904
// MI455X (gfx1250) — compile-verified
//
#include <hip/hip_runtime.h>

// GCN fused layer for MI455X (gfx1250, wave32):
//   out[b,n,o] = relu( (adj[b] @ x[b]) @ kernel + bias )
// B=32, N=1024, F=64, OUT=128.  HBM-bound on adj (134 MB) -> fuse both GEMMs,
// stay in f32 and use V_WMMA_F32_16X16X4_F32.

typedef __attribute__((ext_vector_type(2))) float v2f;
typedef __attribute__((ext_vector_type(8))) float v8f;

#define B_    32
#define N_    1024
#define F_    64
#define OUT_  128
// LDS row stride for the 16x64 agg tile; 68 (not 64) so that phase-2 column
// reads (fixed k, varying row m) hit different banks: bank = (m*68 + k) % 64.
#define LDS_STRIDE 68

__global__ __launch_bounds__(128) void gcn_fused_kernel(
    const float* __restrict__ x,      // [B, N, F]
    const float* __restrict__ adj,    // [B, N, N]
    const float* __restrict__ kern,   // [F, OUT]
    const float* __restrict__ bias,   // [OUT]
    float* __restrict__ out)          // [B, N, OUT]
{
    __shared__ float s_agg[16 * LDS_STRIDE];   // 16 x 64 agg tile (padded)

    const int lane  = threadIdx.x & 31;
    const int wave  = threadIdx.x >> 5;        // 4 waves per block
    const int m     = lane & 15;               // row (A) or column (B, C/D) index
    const int khalf = (lane >> 4) * 2;         // K sub-offset: 0 (lanes 0-15) / 2 (16-31)

    const int b    = blockIdx.y;               // batch
    const int row0 = blockIdx.x * 16;          // N row tile

    const float* adjB = adj + (size_t)b * N_ * N_;
    const float* xB   = x   + (size_t)b * N_ * F_;

    // ---------------- Phase 1: agg tile = adj[rows,:] @ x ----------------
    // Wave `wave` produces agg columns [wave*16, wave*16+16).
    const int colA = wave * 16;

    // A layout (16x4 f32): lane l<16 holds A[m][k+0], A[m][k+1];
    //                      lane l>=16 holds A[m][k+2], A[m][k+3].
    const float* aRow = adjB + (size_t)(row0 + m) * N_ + khalf;
    // B layout (4x16 f32): lane holds B[k+khalf+0][n], B[k+khalf+1][n], n = m.
    const float* bCol = xB + colA + m;

    v8f acc = {};
    #pragma unroll 4
    for (int k = 0; k < N_; k += 4) {
        v2f a = *(const v2f*)(aRow + k);                 // global_load_b64
        v2f bb;
        bb.x = bCol[(k + khalf + 0) * F_];
        bb.y = bCol[(k + khalf + 1) * F_];
        acc = __builtin_amdgcn_wmma_f32_16x16x4_f32(
            /*neg_a=*/false, a, /*neg_b=*/false, bb,
            /*c_mod=*/(short)0, acc, /*reuse_a=*/false, /*reuse_b=*/false);
    }

    // C/D layout: VGPR r -> row r (lanes 0-15) or row r+8 (lanes 16-31), col = lane%16.
    {
        const int rbase = (lane >> 4) * 8;
        #pragma unroll
        for (int r = 0; r < 8; ++r)
            s_agg[(rbase + r) * LDS_STRIDE + colA + m] = acc[r];
    }
    __syncthreads();

    // ------------- Phase 2: out tile = agg(16x64) @ kernel(64x128) -------------
    // 8 output column tiles; each wave does 2.
    #pragma unroll
    for (int t = 0; t < 2; ++t) {
        const int col0 = (wave * 2 + t) * 16;
        v8f acc2 = {};
        #pragma unroll
        for (int k = 0; k < F_; k += 4) {
            // A fragment from LDS agg tile (8-byte aligned: (m*68 + even)*4)
            v2f a = *(const v2f*)(&s_agg[m * LDS_STRIDE + k + khalf]);
            v2f bb;
            bb.x = kern[(k + khalf + 0) * OUT_ + col0 + m];
            bb.y = kern[(k + khalf + 1) * OUT_ + col0 + m];
            acc2 = __builtin_amdgcn_wmma_f32_16x16x4_f32(
                false, a, false, bb, (short)0, acc2, false, false);
        }
        const float bv = bias[col0 + m];
        float* oBase = out + ((size_t)b * N_ + row0 + (lane >> 4) * 8) * OUT_ + col0 + m;
        #pragma unroll
        for (int r = 0; r < 8; ++r) {
            float v = acc2[r] + bv;
            oBase[r * OUT_] = v > 0.0f ? v : 0.0f;   // bias + relu
        }
    }
}

extern "C" void kernel_launch(void* const* d_in, const int* in_sizes, int n_in,
                              void* d_out, int out_size, void* d_ws, size_t ws_size,
                              hipStream_t stream) {
    (void)in_sizes; (void)n_in; (void)d_ws; (void)ws_size; (void)out_size;
    const float* x    = (const float*)d_in[0];   // [32,1024,64]
    const float* adj  = (const float*)d_in[1];   // [32,1024,1024]
    const float* kern = (const float*)d_in[2];   // [64,128]
    const float* bias = (const float*)d_in[3];   // [128]
    float* out = (float*)d_out;                  // [32,1024,128]

    dim3 grid(N_ / 16, B_);   // 64 row tiles x 32 batches
    dim3 block(128);          // 4 wave32s
    gcn_fused_kernel<<<grid, block, 0, stream>>>(x, adj, kern, bias, out);
}